// VQVAE_84112639525588
// MI455X (gfx1250) — compile-verified
//
#include <hip/hip_runtime.h>

typedef __attribute__((ext_vector_type(2))) float v2f;
typedef __attribute__((ext_vector_type(4))) float v4f;
typedef __attribute__((ext_vector_type(8))) float v8f;

#define VQ_B   32
#define VQ_D   64
#define VQ_HW  4096
#define VQ_K   512
#define VQ_N   (VQ_B * VQ_HW)          // 131072 positions
#define VQ_ELEMS (VQ_N * VQ_D)         // 8388608 output elements

#define STAGE_ROWS 128                 // codebook rows staged in LDS per pass
#define ROW_PAD    68                  // 64 + 4 pad floats -> conflict-free ds_load_b64

// ---------------------------------------------------------------------------
// Kernel A: fp32 WMMA distance GEMM + running argmin.
// One wave handles 32 consecutive positions (two 16-row M-tiles sharing every
// B fragment -> half the DS traffic per WMMA, two independent WMMA chains).
// ---------------------------------------------------------------------------
__global__ __launch_bounds__(256) void vq_argmin_kernel(
    const float* __restrict__ feat,   // (B, D, H, W) fp32
    const float* __restrict__ emb,    // (K, D) fp32
    int* __restrict__ idx_out)        // (N,) argmin indices
{
    __shared__ float cb[STAGE_ROWS * ROW_PAD];   // 34,816 B
    __shared__ float e2s[STAGE_ROWS];            //    512 B

    const int tid  = threadIdx.x;
    const int lane = tid & 31;
    const int wave = tid >> 5;
    const int half = lane >> 4;     // 0: lanes 0-15, 1: lanes 16-31
    const int lrow = lane & 15;

    const int n0   = (blockIdx.x * 8 + wave) * 32;   // first position of the two tiles
    const int bpos = n0 >> 12;                        // batch index (HW = 4096)
    const int s0   = n0 & 4095;                       // spatial offset

    // --- Preload A fragments for both M-tiles
    // (ISA f32 16x4 layout: M = lane&15, K = vgpr + 2*(lane>=16))
    // ax*[2*kk + j] = X[row = lrow (+16)][dim = 4*kk + 2*half + j]
    float ax0[32], ax1[32];
    const float* fb = feat + ((size_t)bpos * VQ_D) * VQ_HW + (size_t)(s0 + lrow);
    #pragma unroll
    for (int kk = 0; kk < 16; ++kk) {
        const int d0 = 4 * kk + 2 * half;
        ax0[2 * kk + 0] = fb[(size_t)(d0 + 0) * VQ_HW];
        ax0[2 * kk + 1] = fb[(size_t)(d0 + 1) * VQ_HW];
        ax1[2 * kk + 0] = fb[(size_t)(d0 + 0) * VQ_HW + 16];
        ax1[2 * kk + 1] = fb[(size_t)(d0 + 1) * VQ_HW + 16];
    }

    // Running argmin: lane holds 8 positions per tile (M = r + 8*half), codeword N = lrow
    float bS0[8], bS1[8];
    int   bI0[8], bI1[8];
    #pragma unroll
    for (int r = 0; r < 8; ++r) {
        bS0[r] = 3.4e38f; bI0[r] = 0;
        bS1[r] = 3.4e38f; bI1[r] = 0;
    }

    for (int stage = 0; stage < VQ_K / STAGE_ROWS; ++stage) {
        __syncthreads();   // previous stage fully consumed
        // --- cooperative, coalesced stage of 128 codebook rows into padded LDS
        const float* src = emb + (size_t)stage * (STAGE_ROWS * VQ_D);
        #pragma unroll
        for (int i = 0; i < 8; ++i) {
            const int f   = tid * 4 + i * 1024;        // flat float index in stage
            const v4f  v  = *(const v4f*)(src + f);
            const int row = f >> 6, off = f & 63;
            *(v4f*)(cb + row * ROW_PAD + off) = v;
        }
        __syncthreads();
        // --- per-row |e|^2 (full fp32, consistent across all positions)
        if (tid < STAGE_ROWS) {
            const float* rp = cb + tid * ROW_PAD;
            float s = 0.f;
            #pragma unroll
            for (int d = 0; d < 64; ++d) s = fmaf(rp[d], rp[d], s);
            e2s[tid] = s;
        }
        __syncthreads();

        // --- 8 codeword tiles of 16; 16 WMMA steps x 2 independent chains each
        for (int ct = 0; ct < 8; ++ct) {
            const float* bp  = cb + (ct * 16 + lrow) * ROW_PAD + 2 * half;
            const float  e2v = e2s[ct * 16 + lrow];
            v8f c0 = {}, c1 = {};
            #pragma unroll
            for (int kk = 0; kk < 16; ++kk) {
                const v2f b = *(const v2f*)(bp + 4 * kk);  // ds_load_b64, conflict-free
                v2f a0; a0.x = ax0[2 * kk]; a0.y = ax0[2 * kk + 1];
                v2f a1; a1.x = ax1[2 * kk]; a1.y = ax1[2 * kk + 1];
                c0 = __builtin_amdgcn_wmma_f32_16x16x4_f32(
                         false, a0, false, b, (short)0, c0, false, false);
                c1 = __builtin_amdgcn_wmma_f32_16x16x4_f32(
                         false, a1, false, b, (short)0, c1, false, false);
            }
            const int ci = stage * STAGE_ROWS + ct * 16 + lrow;
            #pragma unroll
            for (int r = 0; r < 8; ++r) {
                const float s0v = fmaf(-2.0f, c0[r], e2v);  // dist - |x|^2 (const per row)
                const float s1v = fmaf(-2.0f, c1[r], e2v);
                if (s0v < bS0[r]) { bS0[r] = s0v; bI0[r] = ci; }  // strict < => first idx wins
                if (s1v < bS1[r]) { bS1[r] = s1v; bI1[r] = ci; }
            }
        }
    }

    // --- reduce across the 16 lanes of each half (masks 1..8 never mix halves)
    #pragma unroll
    for (int m = 1; m <= 8; m <<= 1) {
        #pragma unroll
        for (int r = 0; r < 8; ++r) {
            float os; int oi;
            os = __shfl_xor(bS0[r], m, 32);
            oi = __shfl_xor(bI0[r], m, 32);
            if (os < bS0[r] || (os == bS0[r] && oi < bI0[r])) { bS0[r] = os; bI0[r] = oi; }
            os = __shfl_xor(bS1[r], m, 32);
            oi = __shfl_xor(bI1[r], m, 32);
            if (os < bS1[r] || (os == bS1[r] && oi < bI1[r])) { bS1[r] = os; bI1[r] = oi; }
        }
    }
    if (lrow == 0) {   // lane 0 -> rows 0..7, lane 16 -> rows 8..15 (per tile)
        #pragma unroll
        for (int r = 0; r < 8; ++r) {
            idx_out[n0 + half * 8 + r]      = bI0[r];
            idx_out[n0 + 16 + half * 8 + r] = bI1[r];
        }
    }
}

// ---------------------------------------------------------------------------
// Kernel B: gather codewords, write NCHW output, per-block sum of sq diffs.
// ---------------------------------------------------------------------------
__global__ __launch_bounds__(256) void vq_gather_kernel(
    const float* __restrict__ feat,
    const float* __restrict__ emb,
    const int*   __restrict__ idx,
    float*       __restrict__ out,
    float*       __restrict__ partial)
{
    const int n = blockIdx.x * 256 + threadIdx.x;
    const int b = n >> 12, s = n & 4095;
    const int ci = idx[n];

    const float* ep = emb  + (size_t)ci * VQ_D;
    const float* xp = feat + ((size_t)b * VQ_D) * VQ_HW + s;
    float*       op = out  + ((size_t)b * VQ_D) * VQ_HW + s;

    float acc = 0.f;
    #pragma unroll
    for (int dd = 0; dd < 16; ++dd) {
        const v4f ev = *(const v4f*)(ep + dd * 4);   // codeword: b128 loads (L1/L2 hot)
        #pragma unroll
        for (int j = 0; j < 4; ++j) {
            const int d = dd * 4 + j;
            const float e = ev[j];
            const float x = xp[(size_t)d * VQ_HW];   // coalesced over s
            op[(size_t)d * VQ_HW] = e;               // coalesced over s
            const float df = e - x;
            acc = fmaf(df, df, acc);
        }
    }

    __shared__ float red[256];
    red[threadIdx.x] = acc;
    __syncthreads();
    #pragma unroll
    for (int st = 128; st > 0; st >>= 1) {           // fixed tree: deterministic
        if (threadIdx.x < st) red[threadIdx.x] += red[threadIdx.x + st];
        __syncthreads();
    }
    if (threadIdx.x == 0) partial[blockIdx.x] = red[0];
}

// ---------------------------------------------------------------------------
// Kernel C: deterministic final reduction of 512 partials -> loss scalar.
// ---------------------------------------------------------------------------
__global__ __launch_bounds__(256) void vq_loss_kernel(
    const float* __restrict__ partial, float* __restrict__ loss_out)
{
    __shared__ float red[256];
    red[threadIdx.x] = partial[threadIdx.x] + partial[threadIdx.x + 256];
    __syncthreads();
    #pragma unroll
    for (int st = 128; st > 0; st >>= 1) {
        if (threadIdx.x < st) red[threadIdx.x] += red[threadIdx.x + st];
        __syncthreads();
    }
    if (threadIdx.x == 0)
        loss_out[0] = red[0] * (1.0f + 0.2f) / (float)VQ_ELEMS;  // commit + BETA*codebook
}

// ---------------------------------------------------------------------------
extern "C" void kernel_launch(void* const* d_in, const int* in_sizes, int n_in,
                              void* d_out, int out_size, void* d_ws, size_t ws_size,
                              hipStream_t stream)
{
    (void)in_sizes; (void)n_in; (void)out_size; (void)ws_size;

    const float* feat = (const float*)d_in[0];   // (32,64,64,64) fp32
    const float* emb  = (const float*)d_in[1];   // (512,64) fp32
    float* out = (float*)d_out;                  // st (8388608) ++ loss (1)

    int*   idx_buf = (int*)d_ws;                                 // 131072 ints
    float* partial = (float*)((char*)d_ws + VQ_N * sizeof(int)); // 512 floats

    vq_argmin_kernel<<<VQ_N / 256, 256, 0, stream>>>(feat, emb, idx_buf);
    vq_gather_kernel<<<VQ_N / 256, 256, 0, stream>>>(feat, emb, idx_buf, out, partial);
    vq_loss_kernel<<<1, 256, 0, stream>>>(partial, out + (size_t)VQ_ELEMS);
}